// GINEConvLayer_1494648619556
// MI455X (gfx1250) — compile-verified
//
#include <hip/hip_runtime.h>

// GINEConv layer for MI455X (gfx1250, wave32).
// Phase 1: acc = x + e1[4] + e2[0]                         (self-loop term)
// Phase 2: acc[row] += x[col] + e1[a0] + e2[a1]            (scatter-add, hw f32 atomics)
// Phase 3: out = relu(acc @ W1^T + b1) @ W2^T + b2         (fused dual GEMM, fp32 WMMA)

#define N_NODES 50000
#define N_EDGES 600000
#define DIM     128       // D
#define HID     256       // 2*D
#define AS_STR  (DIM + 4) // padded LDS strides (keep 16B alignment, kill bank conflicts)
#define HS_STR  (HID + 4)

typedef __attribute__((ext_vector_type(2))) float v2f;
typedef __attribute__((ext_vector_type(8))) float v8f;

// ---------------------------------------------------------------- phase 1
__global__ __launch_bounds__(256) void init_acc(const float* __restrict__ x,
                                                const float* __restrict__ e1,
                                                const float* __restrict__ e2,
                                                float* __restrict__ acc) {
  int idx = blockIdx.x * 256 + threadIdx.x;        // exactly N*D threads
  int d   = idx & (DIM - 1);
  acc[idx] = x[idx] + e1[4 * DIM + d] + e2[0 * DIM + d];
}

// ---------------------------------------------------------------- phase 2
__global__ __launch_bounds__(256) void scatter_edges(const float* __restrict__ x,
                                                     const int* __restrict__ edge_index,
                                                     const int* __restrict__ edge_attr,
                                                     const float* __restrict__ e1,
                                                     const float* __restrict__ e2,
                                                     float* __restrict__ acc) {
  int idx = blockIdx.x * 256 + threadIdx.x;        // exactly E*D threads
  int e   = idx >> 7;                              // edge id
  int d   = idx & (DIM - 1);                       // feature id
  int row = edge_index[e];                         // edge_index shape (2,E)
  int col = edge_index[N_EDGES + e];
  int a0  = edge_attr[2 * e];                      // edge_attr shape (E,2)
  int a1  = edge_attr[2 * e + 1];
  float val = x[col * DIM + d] + e1[a0 * DIM + d] + e2[a1 * DIM + d];
  unsafeAtomicAdd(&acc[row * DIM + d], val);       // -> global_atomic_add_f32
}

// ---------------------------------------------------------------- phase 3
// One workgroup = 128 threads (4 wave32) per 16-row tile of nodes.
// Wave fragment layouts per CDNA5 ISA (7.12.2):
//   A 16x4 f32:  lane (m = lane&15), VGPR0 = A[m][k0+2*kh], VGPR1 = A[m][k0+2*kh+1], kh = lane>>4
//   B 4x16 f32:  VGPR j holds row K=k0+j+2*kh, col n = lane&15
//   C/D 16x16:   VGPR j = element [j + 8*kh][n0 + (lane&15)]
__global__ __launch_bounds__(128) void mlp_fused(const float* __restrict__ acc,
                                                 const float* __restrict__ W1,
                                                 const float* __restrict__ b1,
                                                 const float* __restrict__ W2,
                                                 const float* __restrict__ b2,
                                                 float* __restrict__ out) {
  __shared__ float As[16 * AS_STR];   // 8.25 KB  A tile
  __shared__ float Hs[16 * HS_STR];   // 16.25 KB hidden tile
  const int tid  = threadIdx.x;
  const int wave = tid >> 5;
  const int lane = tid & 31;
  const int m    = lane & 15;
  const int kh   = lane >> 4;
  const int row0 = blockIdx.x * 16;   // N = 3125 * 16, no remainder -> EXEC all ones

  // ---- stage A tile (16 x 128) into LDS: thread -> (row = tid>>3, 16 cols)
  {
    const int r  = tid >> 3;
    const int c0 = (tid & 7) * 16;
    const float4* src = (const float4*)(acc + (row0 + r) * DIM + c0);
    float4*       dst = (float4*)(As + r * AS_STR + c0);
#pragma unroll
    for (int i = 0; i < 4; ++i) dst[i] = src[i];
  }
  __syncthreads();

  // ---- H = relu(A @ W1^T + b1): 16 column-tiles, 4 per wave, K = 128
#pragma unroll
  for (int nt = 0; nt < 4; ++nt) {
    const int n0 = (wave * 4 + nt) * 16;
    const float* w1row = W1 + (n0 + m) * DIM;      // B[k][n] = W1[n][k]
    v8f c = {};
#pragma unroll
    for (int k0 = 0; k0 < DIM; k0 += 4) {
      v2f a, b;
      a.x = As[m * AS_STR + k0 + 2 * kh];
      a.y = As[m * AS_STR + k0 + 2 * kh + 1];
      b.x = w1row[k0 + 2 * kh];
      b.y = w1row[k0 + 2 * kh + 1];
      c = __builtin_amdgcn_wmma_f32_16x16x4_f32(false, a, false, b, (short)0, c,
                                                false, false);
    }
    const float bias = b1[n0 + m];
#pragma unroll
    for (int j = 0; j < 8; ++j) {
      const int mj = j + 8 * kh;
      float h = c[j] + bias;
      Hs[mj * HS_STR + n0 + m] = h > 0.0f ? h : 0.0f;
    }
  }
  __syncthreads();

  // ---- Y = H @ W2^T + b2: 8 column-tiles, 2 per wave, K = 256
#pragma unroll
  for (int nt = 0; nt < 2; ++nt) {
    const int n0 = (wave * 2 + nt) * 16;
    const float* w2row = W2 + (n0 + m) * HID;      // B[k][n] = W2[n][k]
    v8f c = {};
#pragma unroll
    for (int k0 = 0; k0 < HID; k0 += 4) {
      v2f a, b;
      a.x = Hs[m * HS_STR + k0 + 2 * kh];
      a.y = Hs[m * HS_STR + k0 + 2 * kh + 1];
      b.x = w2row[k0 + 2 * kh];
      b.y = w2row[k0 + 2 * kh + 1];
      c = __builtin_amdgcn_wmma_f32_16x16x4_f32(false, a, false, b, (short)0, c,
                                                false, false);
    }
    const float bias = b2[n0 + m];
#pragma unroll
    for (int j = 0; j < 8; ++j) {
      const int mj = j + 8 * kh;
      out[(row0 + mj) * DIM + n0 + m] = c[j] + bias;
    }
  }
}

// ---------------------------------------------------------------- launcher
extern "C" void kernel_launch(void* const* d_in, const int* in_sizes, int n_in,
                              void* d_out, int out_size, void* d_ws, size_t ws_size,
                              hipStream_t stream) {
  const float* x          = (const float*)d_in[0];
  const int*   edge_index = (const int*)d_in[1];   // int32 (JAX default x64-off)
  const int*   edge_attr  = (const int*)d_in[2];
  const float* W1         = (const float*)d_in[3];
  const float* b1         = (const float*)d_in[4];
  const float* W2         = (const float*)d_in[5];
  const float* b2         = (const float*)d_in[6];
  const float* e1         = (const float*)d_in[7];
  const float* e2         = (const float*)d_in[8];
  float*       out        = (float*)d_out;
  float*       acc        = (float*)d_ws;          // N*D floats = 25.6 MB scratch

  (void)in_sizes; (void)n_in; (void)out_size; (void)ws_size;

  init_acc<<<(N_NODES * DIM) / 256, 256, 0, stream>>>(x, e1, e2, acc);
  scatter_edges<<<(N_EDGES * DIM) / 256, 256, 0, stream>>>(x, edge_index, edge_attr,
                                                           e1, e2, acc);
  mlp_fused<<<N_NODES / 16, 128, 0, stream>>>(acc, W1, b1, W2, b2, out);
}